// NAFLTEOURS_9139690406425
// MI455X (gfx1250) — compile-verified
//
#include <hip/hip_runtime.h>

typedef __attribute__((ext_vector_type(2))) float v2f;
typedef __attribute__((ext_vector_type(8))) float v8f;

#define W 256
#define WP1 257
#define ROWS 272        // 17 * 16, padded extent for both rows and cols
#define STRIDE 273      // odd stride: bank-conflict-friendly both directions
#define NITER 10

// Everything lives in the log2 domain: exp() -> native v_exp_f32 (exp2),
// log() -> native v_log_f32 (log2). 2w = 512 = 2^9, so the marginals are
// exact: log2(1/512) = -9, log2(1/2) = -1, and log2(2w) = +9.
static constexpr float LOG2E      = 1.4426950408889634f;
static constexpr float LOG2_512   = 9.0f;
static constexpr float NLOG2_512  = -9.0f;
static constexpr float LOG2_HALF  = -1.0f;

__device__ __forceinline__ float fexp2(float x) { return __builtin_amdgcn_exp2f(x); }  // v_exp_f32
__device__ __forceinline__ float flog2(float x) { return __builtin_amdgcn_logf(x); }   // v_log_f32

// One workgroup per (dir, b, h) matrix. The 257x257 sim matrix lives in LDS,
// padded to 272x272 with -inf so every Sinkhorn reduction loop is branch-free:
// exp2(-inf)=0 contributes nothing to the v_wmma_f32_16x16x4_f32 ones-reduction
// (A = ones 16x4  =>  D[m][n] = sum_k B[k][n], K-permutation-invariant).
__global__ __launch_bounds__(256) void sinkhorn_disp_kernel(
    const float* __restrict__ Mr2l,
    const float* __restrict__ Ml2r,
    const float* __restrict__ phi_p,
    float* __restrict__ out)
{
  extern __shared__ float lds[];
  float* sim = lds;                    // ROWS * STRIDE, scaled by log2(e)
  float* uu  = lds + ROWS * STRIDE;    // ROWS (log2 domain; pad entries stay 0)
  float* vv  = uu + ROWS;              // ROWS (log2 domain; pad entries stay 0)

  const int m   = blockIdx.x;          // 0..511
  const int dir = m >> 8;              // 0: right_to_left (causal), 1: transposed mask
  const int bh  = m & 255;             // b*128 + h
  const float phi2 = phi_p[0] * LOG2E;
  const float* __restrict__ M = (dir == 0 ? Mr2l : Ml2r) + (size_t)bh * (W * W);

  const int tid  = threadIdx.x;
  const int lane = tid & 31;
  const int wave = tid >> 5;           // 0..7 (wave32)
  const float NEG_INF = -__builtin_inff();

  // ---- stage sim*log2e (+ mask, phi borders, -inf padding); HBM read once ----
  for (int i = tid; i < ROWS * STRIDE; i += 256) {
    int r = i / STRIDE;
    int c = i - r * STRIDE;
    float val;
    if (r < W && c < W) {
      float x = M[r * W + c] * LOG2E;
      bool masked = (dir == 0) ? (r < c) : (r > c);
      val = masked ? NEG_INF : x;
    } else if (r <= W && c <= W) {
      val = phi2;                      // row/col 256 border
    } else {
      val = NEG_INF;                   // padding: exp2() -> 0
    }
    sim[i] = val;
  }
  for (int i = tid; i < ROWS; i += 256) { uu[i] = 0.0f; vv[i] = 0.0f; }
  __syncthreads();

  v2f ones = {1.0f, 1.0f};
  const int kbase = (lane >> 4) << 1;  // lanes 0-15 -> k0+{0,1}; lanes 16-31 -> k0+{2,3}
  const int nsub  = lane & 15;         // N (position within the 16-wide group)

  for (int it = 0; it < NITER; ++it) {
    // ---- Pass A: v2[c] = log2_nu[c] - log2( sum_r exp2(sim2[r][c] + u2[r]) ) ----
    for (int g = wave; g < 17; g += 8) {
      int col = g * 16 + nsub;                       // 0..271, always in-bounds
      v8f acc = {0.f, 0.f, 0.f, 0.f, 0.f, 0.f, 0.f, 0.f};
      int off = kbase * STRIDE + col;
      for (int k0 = 0; k0 < ROWS; k0 += 4) {         // 68 branch-free iterations
        v2f b;
        b.x = fexp2(sim[off]          + uu[k0 + kbase]);
        b.y = fexp2(sim[off + STRIDE] + uu[k0 + kbase + 1]);
        acc = __builtin_amdgcn_wmma_f32_16x16x4_f32(false, ones, false, b,
                                                    (short)0, acc, false, false);
        off += 4 * STRIDE;
      }
      if (lane < 16 && col < WP1) {                  // D vgpr0 = column sum
        float lnu = (col == W) ? LOG2_HALF : NLOG2_512;
        vv[col] = lnu - flog2(acc[0]);
      }
    }
    __syncthreads();
    // ---- Pass B: u2[r] = log2_mu[r] - log2( sum_c exp2(sim2[r][c] + v2[c]) ) ----
    for (int g = wave; g < 17; g += 8) {
      int row = g * 16 + nsub;
      v8f acc = {0.f, 0.f, 0.f, 0.f, 0.f, 0.f, 0.f, 0.f};
      int off = row * STRIDE + kbase;
      for (int c0 = 0; c0 < ROWS; c0 += 4) {
        v2f b;
        b.x = fexp2(sim[off]     + vv[c0 + kbase]);
        b.y = fexp2(sim[off + 1] + vv[c0 + kbase + 1]);
        acc = __builtin_amdgcn_wmma_f32_16x16x4_f32(false, ones, false, b,
                                                    (short)0, acc, false, false);
        off += 4;
      }
      if (lane < 16 && row < WP1) {
        float lmu = (row == W) ? LOG2_HALF : NLOG2_512;
        uu[row] = lmu - flog2(acc[0]);
      }
    }
    __syncthreads();
  }

  // ---- final disp / occ: one thread per row r of the 256x256 sub-matrix ----
  {
    int r = tid;
    const float* srow = sim + r * STRIDE;
    const float ur = uu[r];
    float best = NEG_INF; int hr = 0;
    for (int c = 0; c < W; ++c) {                    // argmax(attn) == argmax(sim2 + v2)
      float s = srow[c] + vv[c];
      if (s > best) { best = s; hr = c; }
    }
    float a0 = (hr >= 1)    ? fexp2(srow[hr - 1] + ur + vv[hr - 1] + LOG2_512) : 0.0f;
    float a1 =                fexp2(srow[hr]     + ur + vv[hr]     + LOG2_512);
    float a2 = (hr + 1 < W) ? fexp2(srow[hr + 1] + ur + vv[hr + 1] + LOG2_512) : 0.0f;
    float norm  = a0 + a1 + a2;
    float normc = (norm < 0.1f) ? 1.0f : norm;
    float inv   = 1.0f / normc;
    const float scale = 2.0f / 255.0f;               // linspace(0,2,256) spacing
    float p0, p1, p2;
    if (dir == 0) {                                  // pos[r][c] = max(2(r-c)/255, 0)
      p0 = (hr >= 1)    ? fmaxf((float)(r - (hr - 1)) * scale, 0.0f) : 0.0f;
      p1 =                fmaxf((float)(r - hr) * scale, 0.0f);
      p2 = (hr + 1 < W) ? fmaxf((float)(r - (hr + 1)) * scale, 0.0f) : 0.0f;
    } else {                                         // pos.T: max(2(c-r)/255, 0)
      p0 = (hr >= 1)    ? fmaxf((float)((hr - 1) - r) * scale, 0.0f) : 0.0f;
      p1 =                fmaxf((float)(hr - r) * scale, 0.0f);
      p2 = (hr + 1 < W) ? fmaxf((float)((hr + 1) - r) * scale, 0.0f) : 0.0f;
    }
    float disp = (a0 * p0 + a1 * p1 + a2 * p2) * inv;
    float occ  = 1.0f - normc;
    int bb = bh >> 7, hh = bh & 127;
    // out shape (4, 2, 128, 256): [dir*2+0]=disp, [dir*2+1]=occ
    size_t idx_d = ((((size_t)(dir * 2 + 0)) * 2 + bb) * 128 + hh) * 256 + r;
    size_t idx_o = ((((size_t)(dir * 2 + 1)) * 2 + bb) * 128 + hh) * 256 + r;
    out[idx_d] = disp;
    out[idx_o] = occ;
  }
}

extern "C" void kernel_launch(void* const* d_in, const int* in_sizes, int n_in,
                              void* d_out, int out_size, void* d_ws, size_t ws_size,
                              hipStream_t stream) {
  const float* Mr2l = (const float*)d_in[0];
  const float* Ml2r = (const float*)d_in[1];
  const float* phi  = (const float*)d_in[2];
  float* out = (float*)d_out;
  const size_t lds_bytes = (size_t)(ROWS * STRIDE + 2 * ROWS) * sizeof(float); // ~292 KB < 320 KB
  (void)hipFuncSetAttribute((const void*)sinkhorn_disp_kernel,
                            hipFuncAttributeMaxDynamicSharedMemorySize,
                            (int)lds_bytes);
  sinkhorn_disp_kernel<<<dim3(512), dim3(256), lds_bytes, stream>>>(Mr2l, Ml2r, phi, out);
}